// StageOneFitter_57449482551548
// MI455X (gfx1250) — compile-verified
//
#include <hip/hip_runtime.h>
#include <float.h>
#include <math.h>

// ---------------- problem constants ----------------
#define NBATCH 512
#define NMARK  53
#define NVERT  6890
#define NBETA  10
#define NJ     24
#define PFK    207     // pose_feat length
#define PFKP   208     // padded (multiple of 4 for WMMA K-steps)
#define GC     636     // gathered posedirs columns = 53 markers * 4 verts * 3 coords
#define GCP    640     // padded to multiple of 16
#define MT     32      // M tiles (512/16)
#define NT     40      // N tiles (640/16)
#define NPAIRS (NBATCH*NMARK)          // 27136
#define NVG    (NBATCH*NMARK*4)        // 108544
#define NB6    ((NPAIRS + 255)/256)    // 106 partial blocks
#define EPS    1e-8f
#define WT_DATA (75.0f * (46.0f/53.0f))

// ---------------- workspace layout ----------------
// int region at byte 0
#define OFF_IDS   0            // 64 ints
#define OFF_VIDX  256          // 256 ints (53*4 used)
// float region starts at byte 4096; offsets below in floats
#define FO_VS    0             // v_shaped 20670 (+pad)
#define FO_J     20672         // J 24*3
#define FO_MPL   20744         // marker_pose_l 53*4
#define FO_BASE  20956         // base loss (beta + dis)
#define FO_PF    20960         // pose_feat padded 512*208
#define FO_A     127456        // A mats 512*24*12
#define FO_PDG   274912        // gathered posedirs 208*640
#define FO_VP    408032        // gemm out 512*640
#define FO_GV    735712        // gathered verts 512*53*4*3
#define FO_PART  1061344       // 128 partials

__constant__ int c_parents[NJ] = {-1,0,0,0,1,2,3,4,5,6,7,8,9,9,9,12,13,14,16,17,18,19,20,21};

typedef __attribute__((ext_vector_type(2))) float v2f;
typedef __attribute__((ext_vector_type(8))) float v8f;

// ---------------- K0: marker -> nearest template vertex + face verts ----------------
__global__ void k0_argmin(const float* __restrict__ marker_pose,
                          const float* __restrict__ v_template,
                          const int* __restrict__ faces,
                          const int* __restrict__ vertex_faces,
                          int* __restrict__ ids, int* __restrict__ vidx) {
  __shared__ float sb[256];
  __shared__ int   si[256];
  int m = blockIdx.x;
  int t = threadIdx.x;
  float mx = marker_pose[m*3+0], my = marker_pose[m*3+1], mz = marker_pose[m*3+2];
  float best = FLT_MAX; int bidx = 0x7fffffff;
  for (int v = t; v < NVERT; v += 256) {
    float dx = mx - v_template[v*3+0];
    float dy = my - v_template[v*3+1];
    float dz = mz - v_template[v*3+2];
    float d = dx*dx + dy*dy + dz*dz;
    if (d < best) { best = d; bidx = v; }
  }
  sb[t] = best; si[t] = bidx;
  __syncthreads();
  for (int s = 128; s > 0; s >>= 1) {
    if (t < s) {
      float ob = sb[t+s]; int oi = si[t+s];
      if (ob < sb[t] || (ob == sb[t] && oi < si[t])) { sb[t] = ob; si[t] = oi; }
    }
    __syncthreads();
  }
  if (t == 0) {
    int vid = si[0];
    ids[m] = vid;
    int f = vertex_faces[vid];
    vidx[m*4+0] = vid;
    vidx[m*4+1] = faces[f*3+0];
    vidx[m*4+2] = faces[f*3+1];
    vidx[m*4+3] = faces[f*3+2];
  }
}

// ---------------- K1: v_shaped = v_template + shapedirs @ betas ----------------
__global__ void k1_vshaped(const float* __restrict__ v_template,
                           const float* __restrict__ shapedirs,
                           const float* __restrict__ betas,
                           float* __restrict__ vs) {
  int i = blockIdx.x*256 + threadIdx.x;
  if (i >= NVERT*3) return;
  float acc = v_template[i];
  #pragma unroll
  for (int k = 0; k < NBETA; ++k) acc += shapedirs[i*NBETA + k] * betas[k];
  vs[i] = acc;
}

// ---------------- K2: J = J_regressor @ v_shaped ----------------
__global__ void k2_joints(const float* __restrict__ Jr,
                          const float* __restrict__ vs,
                          float* __restrict__ Jw) {
  __shared__ float s0[256], s1[256], s2[256];
  int j = blockIdx.x, t = threadIdx.x;
  float a0 = 0.f, a1 = 0.f, a2 = 0.f;
  for (int v = t; v < NVERT; v += 256) {
    float w = Jr[j*NVERT + v];
    a0 += w * vs[v*3+0]; a1 += w * vs[v*3+1]; a2 += w * vs[v*3+2];
  }
  s0[t] = a0; s1[t] = a1; s2[t] = a2;
  __syncthreads();
  for (int s = 128; s > 0; s >>= 1) {
    if (t < s) { s0[t] += s0[t+s]; s1[t] += s1[t+s]; s2[t] += s2[t+s]; }
    __syncthreads();
  }
  if (t == 0) { Jw[j*3+0] = s0[0]; Jw[j*3+1] = s1[0]; Jw[j*3+2] = s2[0]; }
}

// ---------------- K3: template LCS, marker_pose_l, base loss ----------------
__global__ void k3_lcs(const float* __restrict__ marker_pose,
                       const float* __restrict__ v_template,
                       const float* __restrict__ betas,
                       const int* __restrict__ vidx,
                       float* __restrict__ mpl, float* __restrict__ base) {
  int m = threadIdx.x;
  if (m < NMARK) {
    int v0 = vidx[m*4+0], v1 = vidx[m*4+1], v2 = vidx[m*4+2], v3 = vidx[m*4+3];
    float p[3], a[3], b[3], c[3];
    for (int k = 0; k < 3; ++k) {
      p[k] = v_template[v0*3+k]; a[k] = v_template[v1*3+k];
      b[k] = v_template[v2*3+k]; c[k] = v_template[v3*3+k];
    }
    float e1[3] = {b[0]-a[0], b[1]-a[1], b[2]-a[2]};
    float e2[3] = {c[0]-a[0], c[1]-a[1], c[2]-a[2]};
    float n[3] = {e1[1]*e2[2]-e1[2]*e2[1], e1[2]*e2[0]-e1[0]*e2[2], e1[0]*e2[1]-e1[1]*e2[0]};
    float ln = sqrtf(n[0]*n[0]+n[1]*n[1]+n[2]*n[2]) + EPS;
    n[0] /= ln; n[1] /= ln; n[2] /= ln;
    float le = sqrtf(e1[0]*e1[0]+e1[1]*e1[1]+e1[2]*e1[2]) + EPS;
    float e[3] = {e1[0]/le, e1[1]/le, e1[2]/le};
    float pr = e[0]*n[0]+e[1]*n[1]+e[2]*n[2];
    float y[3] = {e[0]-pr*n[0], e[1]-pr*n[1], e[2]-pr*n[2]};
    float ly = sqrtf(y[0]*y[0]+y[1]*y[1]+y[2]*y[2]) + EPS;
    y[0] /= ly; y[1] /= ly; y[2] /= ly;
    float z[3] = {n[1]*y[2]-n[2]*y[1], n[2]*y[0]-n[0]*y[2], n[0]*y[1]-n[1]*y[0]};
    // lcs_inv @ [mp,1] = R^T (mp - p); R columns = [n y z]
    float d[3] = {marker_pose[m*3+0]-p[0], marker_pose[m*3+1]-p[1], marker_pose[m*3+2]-p[2]};
    mpl[m*4+0] = n[0]*d[0] + n[1]*d[1] + n[2]*d[2];
    mpl[m*4+1] = y[0]*d[0] + y[1]*d[1] + y[2]*d[2];
    mpl[m*4+2] = z[0]*d[0] + z[1]*d[1] + z[2]*d[2];
    mpl[m*4+3] = 1.0f;
  }
  __syncthreads();
  if (m == 0) {
    float lb = 0.f;
    for (int k = 0; k < NBETA; ++k) lb += betas[k]*betas[k];
    float ld = 0.f;
    for (int i = 0; i < NMARK; ++i) {
      float q = mpl[i*4+0] - 0.0095f;
      ld += 10000.0f * q*q;
    }
    base[0] = 10.0f*lb + ld;
  }
}

// ---------------- K3.5: gather padded posedirs columns (208 x 640) ----------------
__global__ void k35_gather_pd(const float* __restrict__ posedirs,
                              const int* __restrict__ vidx,
                              float* __restrict__ pdg) {
  int i = blockIdx.x*256 + threadIdx.x;
  if (i >= PFKP*GCP) return;
  int k = i / GCP, col = i % GCP;
  float val = 0.f;
  if (k < PFK && col < GC) {
    int ms = col / 3, c = col % 3;        // ms = m*4+s
    int v = vidx[ms];
    val = posedirs[(size_t)k*(NVERT*3) + v*3 + c];
  }
  pdg[i] = val;
}

// ---------------- K4: per-batch Rodrigues + chain + A + pose_feat ----------------
__global__ void k4_chain(const float* __restrict__ body_pose,
                         const float* __restrict__ global_orient,
                         const float* __restrict__ Jw,
                         float* __restrict__ pf, float* __restrict__ Aw) {
  int b = blockIdx.x*64 + threadIdx.x;
  if (b >= NBATCH) return;
  float R[NJ][9];
  for (int j = 0; j < NJ; ++j) {
    float ax, ay, az;
    if (j == 0) { ax = global_orient[b*3+0]; ay = global_orient[b*3+1]; az = global_orient[b*3+2]; }
    else { ax = body_pose[b*69 + (j-1)*3 + 0]; ay = body_pose[b*69 + (j-1)*3 + 1]; az = body_pose[b*69 + (j-1)*3 + 2]; }
    float ang = sqrtf(ax*ax + ay*ay + az*az) + EPS;
    float x = ax/ang, y = ay/ang, z = az/ang;
    float s = sinf(ang), cc = cosf(ang), t = 1.0f - cc;
    R[j][0] = 1.f - t*(y*y + z*z); R[j][1] = -s*z + t*x*y;        R[j][2] =  s*y + t*x*z;
    R[j][3] =  s*z + t*x*y;        R[j][4] = 1.f - t*(x*x + z*z); R[j][5] = -s*x + t*y*z;
    R[j][6] = -s*y + t*x*z;        R[j][7] =  s*x + t*y*z;        R[j][8] = 1.f - t*(x*x + y*y);
  }
  // pose_feat (padded to 208)
  for (int j = 1; j < NJ; ++j)
    for (int r = 0; r < 3; ++r)
      for (int c = 0; c < 3; ++c)
        pf[(size_t)b*PFKP + (j-1)*9 + r*3 + c] = R[j][r*3+c] - ((r == c) ? 1.f : 0.f);
  pf[(size_t)b*PFKP + 207] = 0.f;
  // kinematic chain
  float Jl[NJ][3];
  for (int j = 0; j < NJ; ++j) for (int k = 0; k < 3; ++k) Jl[j][k] = Jw[j*3+k];
  float G[NJ][12];
  for (int r = 0; r < 3; ++r) {
    for (int c = 0; c < 3; ++c) G[0][r*4+c] = R[0][r*3+c];
    G[0][r*4+3] = Jl[0][r];
  }
  for (int j = 1; j < NJ; ++j) {
    int par = c_parents[j];
    float rel[3] = {Jl[j][0]-Jl[par][0], Jl[j][1]-Jl[par][1], Jl[j][2]-Jl[par][2]};
    for (int r = 0; r < 3; ++r) {
      for (int c = 0; c < 3; ++c) {
        float acc = 0.f;
        for (int k = 0; k < 3; ++k) acc += G[par][r*4+k] * R[j][k*3+c];
        G[j][r*4+c] = acc;
      }
      float tt = G[par][r*4+3];
      for (int k = 0; k < 3; ++k) tt += G[par][r*4+k] * rel[k];
      G[j][r*4+3] = tt;
    }
  }
  // A = G with t -= R_G * J
  for (int j = 0; j < NJ; ++j)
    for (int r = 0; r < 3; ++r) {
      float* dst = &Aw[(size_t)b*(NJ*12) + j*12 + r*4];
      dst[0] = G[j][r*4+0]; dst[1] = G[j][r*4+1]; dst[2] = G[j][r*4+2];
      dst[3] = G[j][r*4+3] - (G[j][r*4+0]*Jl[j][0] + G[j][r*4+1]*Jl[j][1] + G[j][r*4+2]*Jl[j][2]);
    }
}

// ---------------- K5: WMMA fp32 GEMM: (512x208) @ (208x640) -> vp ----------------
__global__ void k5_gemm(const float* __restrict__ pf,
                        const float* __restrict__ pdg,
                        float* __restrict__ vp) {
  int wid  = (blockIdx.x*256 + threadIdx.x) >> 5;   // 1280 waves total
  int lane = threadIdx.x & 31;
  int tm = wid / NT, tn = wid % NT;
  int rM = lane & 15;           // row within tile for A (M) / col for B (N)
  int kg = lane >> 4;           // K-pair selector
  v8f acc = {0.f,0.f,0.f,0.f,0.f,0.f,0.f,0.f};
  const float* arow = pf  + (size_t)(tm*16 + rM)*PFKP + kg*2;
  const float* bcol = pdg + (size_t)(kg*2)*GCP + tn*16 + rM;
  for (int k0 = 0; k0 < PFKP; k0 += 4) {
    v2f a, b;
    a[0] = arow[k0];            a[1] = arow[k0 + 1];
    b[0] = bcol[(size_t)k0*GCP]; b[1] = bcol[(size_t)(k0+1)*GCP];
    acc = __builtin_amdgcn_wmma_f32_16x16x4_f32(false, a, false, b, (short)0, acc, false, false);
  }
  #pragma unroll
  for (int i = 0; i < 8; ++i)
    vp[(size_t)(tm*16 + i + kg*8)*GCP + tn*16 + rM] = acc[i];
}

// ---------------- K6a: gathered vertices via LBS ----------------
__global__ void k6a_verts(const float* __restrict__ vs,
                          const float* __restrict__ vp,
                          const float* __restrict__ Aw,
                          const float* __restrict__ lbs,
                          const float* __restrict__ transl,
                          const int* __restrict__ vidx,
                          float* __restrict__ gv) {
  int idx = blockIdx.x*256 + threadIdx.x;
  if (idx >= NVG) return;
  int b = idx / (NMARK*4);
  int rem = idx % (NMARK*4);      // rem = m*4+s
  int v = vidx[rem];
  float vp0 = vs[v*3+0] + vp[(size_t)b*GCP + rem*3 + 0];
  float vp1 = vs[v*3+1] + vp[(size_t)b*GCP + rem*3 + 1];
  float vp2 = vs[v*3+2] + vp[(size_t)b*GCP + rem*3 + 2];
  float T[12];
  #pragma unroll
  for (int i = 0; i < 12; ++i) T[i] = 0.f;
  const float* Ab = Aw + (size_t)b*(NJ*12);
  for (int j = 0; j < NJ; ++j) {
    float w = lbs[(size_t)v*NJ + j];
    const float* Aj = Ab + j*12;
    #pragma unroll
    for (int i = 0; i < 12; ++i) T[i] += w * Aj[i];
  }
  #pragma unroll
  for (int r = 0; r < 3; ++r)
    gv[(size_t)idx*3 + r] =
        T[r*4+0]*vp0 + T[r*4+1]*vp1 + T[r*4+2]*vp2 + T[r*4+3] + transl[b*3+r];
}

// ---------------- K6b: posed LCS + data loss partials ----------------
__global__ void k6b_loss(const float* __restrict__ gv,
                         const float* __restrict__ mpl,
                         const float* __restrict__ train_data,
                         float* __restrict__ part) {
  __shared__ float sm[256];
  int idx = blockIdx.x*256 + threadIdx.x;
  float local = 0.f;
  if (idx < NPAIRS) {
    int b = idx / NMARK, m = idx % NMARK;
    const float* base = gv + (size_t)(b*NMARK + m)*12;
    const float *p = base, *a = base+3, *bb = base+6, *cv = base+9;
    float e1[3] = {bb[0]-a[0], bb[1]-a[1], bb[2]-a[2]};
    float e2[3] = {cv[0]-a[0], cv[1]-a[1], cv[2]-a[2]};
    float n[3] = {e1[1]*e2[2]-e1[2]*e2[1], e1[2]*e2[0]-e1[0]*e2[2], e1[0]*e2[1]-e1[1]*e2[0]};
    float ln = sqrtf(n[0]*n[0]+n[1]*n[1]+n[2]*n[2]) + EPS;
    n[0]/=ln; n[1]/=ln; n[2]/=ln;
    float le = sqrtf(e1[0]*e1[0]+e1[1]*e1[1]+e1[2]*e1[2]) + EPS;
    float e[3] = {e1[0]/le, e1[1]/le, e1[2]/le};
    float pr = e[0]*n[0]+e[1]*n[1]+e[2]*n[2];
    float y[3] = {e[0]-pr*n[0], e[1]-pr*n[1], e[2]-pr*n[2]};
    float ly = sqrtf(y[0]*y[0]+y[1]*y[1]+y[2]*y[2]) + EPS;
    y[0]/=ly; y[1]/=ly; y[2]/=ly;
    float z[3] = {n[1]*y[2]-n[2]*y[1], n[2]*y[0]-n[0]*y[2], n[0]*y[1]-n[1]*y[0]};
    float l0 = mpl[m*4+0], l1 = mpl[m*4+1], l2 = mpl[m*4+2];
    for (int r = 0; r < 3; ++r) {
      float mg = n[r]*l0 + y[r]*l1 + z[r]*l2 + p[r];
      float d = mg - train_data[(size_t)(b*NMARK + m)*3 + r];
      local += d*d;
    }
    local *= WT_DATA;
  }
  int t = threadIdx.x;
  sm[t] = local;
  __syncthreads();
  for (int s = 128; s > 0; s >>= 1) {
    if (t < s) sm[t] += sm[t+s];
    __syncthreads();
  }
  if (t == 0) part[blockIdx.x] = sm[0];
}

// ---------------- K7: deterministic final sum ----------------
__global__ void k7_final(const float* __restrict__ part,
                         const float* __restrict__ base,
                         float* __restrict__ out) {
  if (threadIdx.x == 0 && blockIdx.x == 0) {
    float s = base[0];
    for (int i = 0; i < NB6; ++i) s += part[i];
    out[0] = s;
  }
}

extern "C" void kernel_launch(void* const* d_in, const int* in_sizes, int n_in,
                              void* d_out, int out_size, void* d_ws, size_t ws_size,
                              hipStream_t stream) {
  const float* train_data    = (const float*)d_in[0];
  const float* marker_pose   = (const float*)d_in[1];
  const float* betas         = (const float*)d_in[2];
  const float* body_pose     = (const float*)d_in[3];
  const float* global_orient = (const float*)d_in[4];
  const float* transl        = (const float*)d_in[5];
  const float* v_template    = (const float*)d_in[6];
  const float* shapedirs     = (const float*)d_in[7];
  const float* posedirs      = (const float*)d_in[8];
  const float* J_regressor   = (const float*)d_in[9];
  const float* lbs_weights   = (const float*)d_in[10];
  const int*   faces         = (const int*)d_in[11];
  const int*   vertex_faces  = (const int*)d_in[12];

  char* wsb = (char*)d_ws;
  int*   ids  = (int*)(wsb + OFF_IDS);
  int*   vidx = (int*)(wsb + OFF_VIDX);
  float* fw   = (float*)(wsb + 4096);
  float* vs   = fw + FO_VS;
  float* Jw   = fw + FO_J;
  float* mpl  = fw + FO_MPL;
  float* base = fw + FO_BASE;
  float* pf   = fw + FO_PF;
  float* Aw   = fw + FO_A;
  float* pdg  = fw + FO_PDG;
  float* vp   = fw + FO_VP;
  float* gv   = fw + FO_GV;
  float* part = fw + FO_PART;

  k0_argmin<<<NMARK, 256, 0, stream>>>(marker_pose, v_template, faces, vertex_faces, ids, vidx);
  k1_vshaped<<<(NVERT*3 + 255)/256, 256, 0, stream>>>(v_template, shapedirs, betas, vs);
  k2_joints<<<NJ, 256, 0, stream>>>(J_regressor, vs, Jw);
  k3_lcs<<<1, 64, 0, stream>>>(marker_pose, v_template, betas, vidx, mpl, base);
  k35_gather_pd<<<(PFKP*GCP + 255)/256, 256, 0, stream>>>(posedirs, vidx, pdg);
  k4_chain<<<(NBATCH + 63)/64, 64, 0, stream>>>(body_pose, global_orient, Jw, pf, Aw);
  k5_gemm<<<(MT*NT)/8, 256, 0, stream>>>(pf, pdg, vp);
  k6a_verts<<<(NVG + 255)/256, 256, 0, stream>>>(vs, vp, Aw, lbs_weights, transl, vidx, gv);
  k6b_loss<<<NB6, 256, 0, stream>>>(gv, mpl, train_data, part);
  k7_final<<<1, 32, 0, stream>>>(part, base, (float*)d_out);
}